// CodeGCN_90202903150609
// MI455X (gfx1250) — compile-verified
//
#include <hip/hip_runtime.h>
#include <hip/hip_bf16.h>

#define NN 100000
#define NE 1000000
#define NG 256
#define ED 64
#define HD 32

typedef float v2f __attribute__((ext_vector_type(2)));
typedef float v8f __attribute__((ext_vector_type(8)));

// ---------- utility fills ----------
__global__ void fill_kernel(float* p, int n, float v) {
    int i = blockIdx.x * blockDim.x + threadIdx.x;
    if (i < n) p[i] = v;
}

// ---------- degree (self-loop baked in via init=1.0) ----------
__global__ void degree_kernel(const int* __restrict__ dst, float* __restrict__ deg) {
    int e = blockIdx.x * blockDim.x + threadIdx.x;
    if (e < NE) atomicAdd(&deg[dst[e]], 1.0f);
}

__global__ void dinv_kernel(float* __restrict__ deg) {
    int i = blockIdx.x * blockDim.x + threadIdx.x;
    if (i < NN) deg[i] = rsqrtf(deg[i]);   // deg >= 1 always (self loop)
}

// ---------- GEMM1: xw = emb[ids] @ W1  (fp32 WMMA 16x16x4) ----------
// One wave -> 16 rows x 32 cols. A: v2f/lane; B/D row-striped across lanes.
__global__ void gemm_emb_wmma(const int* __restrict__ ids,
                              const float* __restrict__ emb,
                              const float* __restrict__ W,
                              float* __restrict__ out) {
    int gwave = (blockIdx.x * blockDim.x + threadIdx.x) >> 5;
    int lane  = threadIdx.x & 31;
    int base  = gwave * 16;
    if (base >= NN) return;                 // wave-uniform: EXEC stays all-1s

    int r16 = lane & 15;
    int kh  = lane >> 4;                    // 0: K lanes {0,1}, 1: K lanes {2,3}
    int arow_idx = base + r16;
    if (arow_idx >= NN) arow_idx = NN - 1;  // clamp gather; masked at store
    const float* arow = emb + (long long)ids[arow_idx] * ED;

    v8f c0 = {}; v8f c1 = {};
    #pragma unroll
    for (int kk = 0; kk < ED / 4; ++kk) {
        int kb = kk * 4 + 2 * kh;
        v2f a;  a.x = arow[kb];             a.y = arow[kb + 1];
        v2f b0; b0.x = W[kb * HD + r16];    b0.y = W[(kb + 1) * HD + r16];
        v2f b1; b1.x = W[kb * HD + r16+16]; b1.y = W[(kb + 1) * HD + r16+16];
        c0 = __builtin_amdgcn_wmma_f32_16x16x4_f32(false, a, false, b0, (short)0, c0, false, false);
        c1 = __builtin_amdgcn_wmma_f32_16x16x4_f32(false, a, false, b1, (short)0, c1, false, false);
    }
    #pragma unroll
    for (int r = 0; r < 8; ++r) {
        int row = base + r + 8 * kh;        // D: vgpr r holds M=r (lanes0-15), M=r+8 (lanes16-31)
        if (row < NN) {
            out[row * HD + r16]      = c0[r];
            out[row * HD + r16 + 16] = c1[r];
        }
    }
}

// ---------- GEMM2: xw = h1 @ W2  (K = 32) ----------
__global__ void gemm_h_wmma(const float* __restrict__ h,
                            const float* __restrict__ W,
                            float* __restrict__ out) {
    int gwave = (blockIdx.x * blockDim.x + threadIdx.x) >> 5;
    int lane  = threadIdx.x & 31;
    int base  = gwave * 16;
    if (base >= NN) return;

    int r16 = lane & 15;
    int kh  = lane >> 4;
    int arow_idx = base + r16;
    if (arow_idx >= NN) arow_idx = NN - 1;
    const float* arow = h + (long long)arow_idx * HD;

    v8f c0 = {}; v8f c1 = {};
    #pragma unroll
    for (int kk = 0; kk < HD / 4; ++kk) {
        int kb = kk * 4 + 2 * kh;
        v2f a;  a.x = arow[kb];             a.y = arow[kb + 1];
        v2f b0; b0.x = W[kb * HD + r16];    b0.y = W[(kb + 1) * HD + r16];
        v2f b1; b1.x = W[kb * HD + r16+16]; b1.y = W[(kb + 1) * HD + r16+16];
        c0 = __builtin_amdgcn_wmma_f32_16x16x4_f32(false, a, false, b0, (short)0, c0, false, false);
        c1 = __builtin_amdgcn_wmma_f32_16x16x4_f32(false, a, false, b1, (short)0, c1, false, false);
    }
    #pragma unroll
    for (int r = 0; r < 8; ++r) {
        int row = base + r + 8 * kh;
        if (row < NN) {
            out[row * HD + r16]      = c0[r];
            out[row * HD + r16 + 16] = c1[r];
        }
    }
}

// ---------- edge scatter: acc[dst] += xw[src] * dinv[src]*dinv[dst] ----------
__global__ void scatter_kernel(const int* __restrict__ src, const int* __restrict__ dst,
                               const float* __restrict__ xw, const float* __restrict__ dinv,
                               float* __restrict__ acc) {
    unsigned int t = blockIdx.x * blockDim.x + threadIdx.x;
    unsigned int e = t >> 5;                // one wave per edge
    unsigned int c = t & 31;                // one lane per feature
    if (e >= NE) return;
    int s = src[e], d = dst[e];
    float norm = dinv[s] * dinv[d];
    atomicAdd(&acc[(long long)d * HD + c], xw[(long long)s * HD + c] * norm);
}

// ---------- finalize: h = relu(acc + self_loop + bias), in place ----------
__global__ void finalize_kernel(const float* __restrict__ xw, const float* __restrict__ dinv,
                                const float* __restrict__ b, float* __restrict__ acc) {
    int t = blockIdx.x * blockDim.x + threadIdx.x;
    if (t >= NN * HD) return;
    int n = t >> 5, c = t & 31;
    float di = dinv[n];
    float v = acc[t] + xw[t] * di * di + b[c];
    acc[t] = v > 0.0f ? v : 0.0f;
}

// ---------- mean pooling by graph ----------
__global__ void pool_kernel(const int* __restrict__ batch, const float* __restrict__ h,
                            float* __restrict__ pooled, float* __restrict__ counts) {
    int t = blockIdx.x * blockDim.x + threadIdx.x;
    if (t >= NN * HD) return;
    int n = t >> 5, c = t & 31;
    int g = batch[n];
    atomicAdd(&pooled[g * HD + c], h[t]);
    if (c == 0) atomicAdd(&counts[g], 1.0f);
}

// ---------- MLP head: sigmoid(relu(pooled @ Wc1 + bc1) @ Wc2 + bc2) ----------
__global__ void head_kernel(const float* __restrict__ pooled, const float* __restrict__ counts,
                            const float* __restrict__ Wc1, const float* __restrict__ bc1,
                            const float* __restrict__ Wc2, const float* __restrict__ bc2,
                            float* __restrict__ out) {
    int g = blockIdx.x * blockDim.x + threadIdx.x;
    if (g >= NG) return;
    float inv = 1.0f / fmaxf(counts[g], 1.0f);
    float pm[HD];
    #pragma unroll
    for (int c = 0; c < HD; ++c) pm[c] = pooled[g * HD + c] * inv;
    float z = bc2[0];
    #pragma unroll
    for (int j = 0; j < 16; ++j) {
        float hj = bc1[j];
        #pragma unroll
        for (int c = 0; c < HD; ++c) hj += pm[c] * Wc1[c * 16 + j];
        hj = fmaxf(hj, 0.0f);
        z += hj * Wc2[j];
    }
    out[g] = 1.0f / (1.0f + expf(-z));
}

extern "C" void kernel_launch(void* const* d_in, const int* in_sizes, int n_in,
                              void* d_out, int out_size, void* d_ws, size_t ws_size,
                              hipStream_t stream) {
    const int*   ids   = (const int*)d_in[0];
    const int*   ei    = (const int*)d_in[1];
    const int*   batch = (const int*)d_in[2];
    const float* emb   = (const float*)d_in[3];
    const float* W1    = (const float*)d_in[4];
    const float* b1    = (const float*)d_in[5];
    const float* W2    = (const float*)d_in[6];
    const float* b2    = (const float*)d_in[7];
    const float* Wc1   = (const float*)d_in[8];
    const float* bc1   = (const float*)d_in[9];
    const float* Wc2   = (const float*)d_in[10];
    const float* bc2   = (const float*)d_in[11];
    float* out = (float*)d_out;

    float* ws     = (float*)d_ws;
    float* deg    = ws;                        // N (becomes dinv in place)
    float* xw     = deg + NN;                  // 32N
    float* acc    = xw + (size_t)NN * HD;      // 32N (reused as h1, then acc2/h2)
    float* pooled = acc + (size_t)NN * HD;     // 256*32
    float* counts = pooled + NG * HD;          // 256

    const int* esrc = ei;
    const int* edst = ei + NE;

    const int B = 256;
    int gNodes   = (NN + B - 1) / B;
    int gFeat    = (NN * HD + B - 1) / B;
    int gEdges   = (NE + B - 1) / B;
    int gScatter = (int)(((long long)NE * 32 + B - 1) / B);
    int gGemm    = (((NN + 15) / 16) + 7) / 8;   // 8 waves/block, 16 rows/wave

    // init: deg=1 (self loop), acc=0, pooled+counts=0
    fill_kernel<<<gNodes, B, 0, stream>>>(deg, NN, 1.0f);
    fill_kernel<<<gFeat,  B, 0, stream>>>(acc, NN * HD, 0.0f);
    fill_kernel<<<(NG * HD + NG + B - 1) / B, B, 0, stream>>>(pooled, NG * HD + NG, 0.0f);

    // degrees -> dinv
    degree_kernel<<<gEdges, B, 0, stream>>>(edst, deg);
    dinv_kernel<<<gNodes, B, 0, stream>>>(deg);

    // layer 1
    gemm_emb_wmma<<<gGemm, B, 0, stream>>>(ids, emb, W1, xw);
    scatter_kernel<<<gScatter, B, 0, stream>>>(esrc, edst, xw, deg, acc);
    finalize_kernel<<<gFeat, B, 0, stream>>>(xw, deg, b1, acc);   // acc = h1

    // layer 2
    gemm_h_wmma<<<gGemm, B, 0, stream>>>(acc, W2, xw);            // xw = h1 @ W2
    fill_kernel<<<gFeat, B, 0, stream>>>(acc, NN * HD, 0.0f);     // acc reused as acc2
    scatter_kernel<<<gScatter, B, 0, stream>>>(esrc, edst, xw, deg, acc);
    finalize_kernel<<<gFeat, B, 0, stream>>>(xw, deg, b2, acc);   // acc = h2

    // pooling + head
    pool_kernel<<<gFeat, B, 0, stream>>>(batch, acc, pooled, counts);
    head_kernel<<<1, NG, 0, stream>>>(pooled, counts, Wc1, bc1, Wc2, bc2, out);
}